// MOEGate_39582418600421
// MI455X (gfx1250) — compile-verified
//
#include <hip/hip_runtime.h>
#include <math.h>

#define TOP_K 2
#define N_EXPERTS 64
#define DIM 2048
#define ALPHA 0.1f
#define BATCH 4
#define SEQ 4096
#define TOKENS (BATCH * SEQ)

#define TOK_PER_WAVE 16
#define WAVES_PER_BLOCK 8
#define TOK_PER_BLOCK (TOK_PER_WAVE * WAVES_PER_BLOCK)   // 128
#define NUM_BLOCKS (TOKENS / TOK_PER_BLOCK)              // 128

typedef float v2f __attribute__((ext_vector_type(2)));
typedef float v8f __attribute__((ext_vector_type(8)));

// ---------------------------------------------------------------------------
// Zero the per-batch accumulation workspace (score sums + top-k counts).
// ws layout: [0, 256) float = sum of softmax probs per (batch, expert)
//            [256, 512) float = top-k selection counts per (batch, expert)
// ---------------------------------------------------------------------------
__global__ void __launch_bounds__(512, 1)
moe_gate_init_kernel(float* __restrict__ ws) {
    int i = blockIdx.x * blockDim.x + threadIdx.x;
    if (i < 2 * BATCH * N_EXPERTS) ws[i] = 0.0f;
}

// ---------------------------------------------------------------------------
// Main: fp32 WMMA GEMM (logits) + softmax + top-2 + per-block aux reductions.
// One wave handles 16 tokens x 64 experts with V_WMMA_F32_16X16X4_F32.
// ---------------------------------------------------------------------------
__global__ void __launch_bounds__(256, 1)
moe_gate_main_kernel(const float* __restrict__ x,    // [TOKENS, DIM]
                     const float* __restrict__ w,    // [N_EXPERTS, DIM]
                     float* __restrict__ out,        // indices | scores | aux
                     float* __restrict__ ws)         // [2, BATCH, N_EXPERTS]
{
    __shared__ float sLogit[TOK_PER_BLOCK][N_EXPERTS];  // 32 KB
    __shared__ float sCnt[N_EXPERTS];

    const int tid   = threadIdx.x;
    const int wave  = tid >> 5;
    const int lane  = tid & 31;
    const int nIdx  = lane & 15;         // token row (A) / expert col (B)
    const int khalf = lane >> 4;         // K half within the 16x16x4 fragment
    const int tBase = blockIdx.x * TOK_PER_BLOCK + wave * TOK_PER_WAVE;
    const int batch = (blockIdx.x * TOK_PER_BLOCK) / SEQ;

    if (tid < N_EXPERTS) sCnt[tid] = 0.0f;

    // ---- WMMA GEMM: logits[16 tokens][64 experts] -------------------------
    // A (16x4 f32): lanes 0-15 rows M=0..15 K={k,k+1}; lanes 16-31 K={k+2,k+3}
    // B (4x16 f32): lanes 0-15 cols N=0..15 K={k,k+1}; lanes 16-31 K={k+2,k+3}
    const float* aRow = x + (size_t)(tBase + nIdx) * DIM + 2 * khalf;
    const float* wRow = w + (size_t)nIdx * DIM + 2 * khalf;
    const float* b0 = wRow;                          // experts  0..15
    const float* b1 = wRow + (size_t)16 * DIM;       // experts 16..31
    const float* b2 = wRow + (size_t)32 * DIM;       // experts 32..47
    const float* b3 = wRow + (size_t)48 * DIM;       // experts 48..63

    v8f acc0 = {}, acc1 = {}, acc2 = {}, acc3 = {};

#pragma unroll 4
    for (int k = 0; k < DIM; k += 4) {
        v2f a  = *(const v2f*)(aRow + k);
        v2f w0 = *(const v2f*)(b0 + k);
        v2f w1 = *(const v2f*)(b1 + k);
        v2f w2 = *(const v2f*)(b2 + k);
        v2f w3 = *(const v2f*)(b3 + k);
        acc0 = __builtin_amdgcn_wmma_f32_16x16x4_f32(false, a, false, w0,
                                                     (short)0, acc0, false, false);
        acc1 = __builtin_amdgcn_wmma_f32_16x16x4_f32(false, a, false, w1,
                                                     (short)0, acc1, false, false);
        acc2 = __builtin_amdgcn_wmma_f32_16x16x4_f32(false, a, false, w2,
                                                     (short)0, acc2, false, false);
        acc3 = __builtin_amdgcn_wmma_f32_16x16x4_f32(false, a, false, w3,
                                                     (short)0, acc3, false, false);
    }

    // ---- Spill C/D fragments to LDS as [token][expert] --------------------
    // C layout: VGPR v -> row (v + 8*khalf), lane%16 -> column.
    {
        const int rBase = wave * TOK_PER_WAVE + khalf * 8;
#pragma unroll
        for (int v = 0; v < 8; ++v) {
            sLogit[rBase + v][nIdx     ] = acc0[v];
            sLogit[rBase + v][nIdx + 16] = acc1[v];
            sLogit[rBase + v][nIdx + 32] = acc2[v];
            sLogit[rBase + v][nIdx + 48] = acc3[v];
        }
    }
    __syncthreads();

    // ---- Softmax + top-2 per token (lanes 0..15 of each wave) -------------
    if (lane < 16) {
        const int tLocal = wave * TOK_PER_WAVE + lane;
        float lg[N_EXPERTS];
        const float4* rp = (const float4*)&sLogit[tLocal][0];
#pragma unroll
        for (int i = 0; i < N_EXPERTS / 4; ++i) {
            float4 q = rp[i];
            lg[4 * i + 0] = q.x; lg[4 * i + 1] = q.y;
            lg[4 * i + 2] = q.z; lg[4 * i + 3] = q.w;
        }

        float mx = -INFINITY;
        float v1 = -INFINITY, v2 = -INFINITY;
        int   i1 = 0, i2 = 0;
#pragma unroll
        for (int e = 0; e < N_EXPERTS; ++e) {
            float t = lg[e];
            mx = fmaxf(mx, t);
            if (t > v1)      { v2 = v1; i2 = i1; v1 = t; i1 = e; }
            else if (t > v2) { v2 = t;  i2 = e; }
        }

        float s = 0.0f;
#pragma unroll
        for (int e = 0; e < N_EXPERTS; ++e) {
            float ee = expf(lg[e] - mx);
            lg[e] = ee;
            s += ee;
        }
        const float inv = 1.0f / s;

        // write softmax probs back for the per-expert column reduction
        float4* wp = (float4*)&sLogit[tLocal][0];
#pragma unroll
        for (int i = 0; i < N_EXPERTS / 4; ++i) {
            float4 q;
            q.x = lg[4 * i + 0] * inv; q.y = lg[4 * i + 1] * inv;
            q.z = lg[4 * i + 2] * inv; q.w = lg[4 * i + 3] * inv;
            wp[i] = q;
        }

        const float p1  = expf(v1 - mx) * inv;
        const float p2  = expf(v2 - mx) * inv;
        const float den = p1 + p2 + 1e-20f;

        const size_t t = (size_t)blockIdx.x * TOK_PER_BLOCK + tLocal;
        out[2 * t + 0] = (float)i1;                     // topk_index
        out[2 * t + 1] = (float)i2;
        out[2 * (size_t)TOKENS + 2 * t + 0] = p1 / den; // topk_scores (renorm)
        out[2 * (size_t)TOKENS + 2 * t + 1] = p2 / den;

        atomicAdd(&sCnt[i1], 1.0f);
        atomicAdd(&sCnt[i2], 1.0f);
    }
    __syncthreads();

    // ---- Per-block column reduction -> one global atomic per expert -------
    if (tid < N_EXPERTS) {
        float colSum = 0.0f;
        for (int r = 0; r < TOK_PER_BLOCK; ++r) colSum += sLogit[r][tid];
        atomicAdd(ws + batch * N_EXPERTS + tid, colSum);
        atomicAdd(ws + BATCH * N_EXPERTS + batch * N_EXPERTS + tid, sCnt[tid]);
    }
}

// ---------------------------------------------------------------------------
// Finalize: aux_loss = ALPHA * mean_b sum_e ce[b,e] * mean_s(scores)[b,e]
// ce[b,e] = count[b,e] * E / (S * K)
// ---------------------------------------------------------------------------
__global__ void __launch_bounds__(64, 1)
moe_gate_finalize_kernel(const float* __restrict__ ws, float* __restrict__ out) {
    __shared__ float red[N_EXPERTS];
    const int e = threadIdx.x;
    float a = 0.0f;
    for (int b = 0; b < BATCH; ++b) {
        float psum = ws[b * N_EXPERTS + e];
        float cnt  = ws[BATCH * N_EXPERTS + b * N_EXPERTS + e];
        float ce   = cnt * ((float)N_EXPERTS / (float)(SEQ * TOP_K));
        a += ce * (psum / (float)SEQ);
    }
    red[e] = a;
    __syncthreads();
    if (e == 0) {
        float tot = 0.0f;
        for (int i = 0; i < N_EXPERTS; ++i) tot += red[i];
        out[(size_t)2 * TOKENS * 2] = tot * (ALPHA / (float)BATCH);
    }
}

// ---------------------------------------------------------------------------
extern "C" void kernel_launch(void* const* d_in, const int* in_sizes, int n_in,
                              void* d_out, int out_size, void* d_ws, size_t ws_size,
                              hipStream_t stream) {
    const float* x = (const float*)d_in[0];   // hidden_states [4,4096,2048]
    const float* w = (const float*)d_in[1];   // weight [64,2048]
    float* out = (float*)d_out;               // idx(32768) | scores(32768) | aux(1)
    float* ws  = (float*)d_ws;                // 512 floats

    moe_gate_init_kernel<<<1, 512, 0, stream>>>(ws);
    moe_gate_main_kernel<<<NUM_BLOCKS, 256, 0, stream>>>(x, w, out, ws);
    moe_gate_finalize_kernel<<<1, 64, 0, stream>>>(ws, out);
}